// DS_CycleGCNPRedictor_63969242907022
// MI455X (gfx1250) — compile-verified
//
#include <hip/hip_runtime.h>
#include <hip/hip_bf16.h>

// ---------------------------------------------------------------------------
// MI455X / gfx1250 implementation of the 3-surface cyclic GCN predictor.
//
// Live dataflow only (dead branches of the reference eliminated):
//   A2 = relu(gcn(x2,Wl2,bl2,ei2));  A1 = relu(gcn(x1,Wl1,bl1,ei1))
//   L1 = A1 + scatter_mean(A2[le2[1]] -> le2[0])
//   LF2= relu(gcn(A2,Wo2,bo2,ei2));  F  = LF2 + scatter_mean(L1[le2[0]] -> le2[1])
//   H  = relu(gcn(F,Wp1,bp1,ei2));   out= gcn(H,Wp2,bp2,ei2)
//
// Dense parts use V_WMMA_F32_16X16X4_F32 (fp32-exact matrix pipe, wave32:
// one wave produces a 16x32 tile). Sparse parts are edge-parallel coalesced
// gathers + global_atomic_add_f32 scatters (the true bandwidth bound here).
// ---------------------------------------------------------------------------

typedef __attribute__((ext_vector_type(2))) float v2f;
typedef __attribute__((ext_vector_type(8))) float v8f;

#define FIN 20
#define DIM 32

// ---------------- utility kernels ----------------

__global__ void zero_f32(float* __restrict__ p, long long n) {
  long long t = (long long)blockIdx.x * blockDim.x + threadIdx.x;
  if (t < n) p[t] = 0.0f;
}

// deg[idx[e]] += 1  (used both for GCN degree and scatter_mean counts)
__global__ void count_idx(float* __restrict__ deg, const int* __restrict__ idx, int m) {
  int t = blockIdx.x * blockDim.x + threadIdx.x;
  if (t < m) atomicAdd(deg + idx[t], 1.0f);
}

// deg -> rsqrt(deg + 1)  (self loop adds 1)
__global__ void finalize_dinv(float* __restrict__ deg, int n) {
  int t = blockIdx.x * blockDim.x + threadIdx.x;
  if (t < n) deg[t] = rsqrtf(deg[t] + 1.0f);
}

// ---------------- dense: Y[N x 32] = X[N x K] * W[K x 32] via WMMA f32 ----------------

template <int KDIM>
__global__ __launch_bounds__(256) void gemm_wmma_k(const float* __restrict__ X,
                                                   const float* __restrict__ W,
                                                   float* __restrict__ Y, int nrows) {
  // one wave32 -> one 16x32 output tile; two 16x16 WMMA column tiles
  int wave = (int)((blockIdx.x * (unsigned)blockDim.x + threadIdx.x) >> 5);
  int lane = threadIdx.x & 31;
  int row0 = wave * 16;
  if (row0 >= nrows) return;          // wave-uniform exit: EXEC all-1 inside
  int lh = lane >> 4;                 // 0: K pair {0,1}; 1: K pair {2,3}
  int lm = lane & 15;                 // M (for A) / N (for B)

  v8f c0 = {};
  v8f c1 = {};
  for (int k0 = 0; k0 < KDIM; k0 += 4) {
    // A tile 16x4: lanes 0-15 rows M=lane K={k0,k0+1}; lanes 16-31 K={k0+2,k0+3}
    const float* xp = X + (size_t)(row0 + lm) * KDIM + (k0 + lh * 2);
    v2f a;
    a.x = xp[0];
    a.y = xp[1];
    // B tile 4x16 (col tiles 0 and 1), rows of W striped across lanes
    const float* wp = W + (size_t)(k0 + lh * 2) * DIM + lm;
    v2f b0, b1;
    b0.x = wp[0];        b0.y = wp[DIM];
    b1.x = wp[16];       b1.y = wp[16 + DIM];
    c0 = __builtin_amdgcn_wmma_f32_16x16x4_f32(false, a, false, b0, (short)0, c0, false, false);
    c1 = __builtin_amdgcn_wmma_f32_16x16x4_f32(false, a, false, b1, (short)0, c1, false, false);
  }
  // C/D layout: VGPR r holds M=r (lanes 0-15) / M=r+8 (lanes 16-31), N=lm
  float* yp = Y + (size_t)row0 * DIM;
#pragma unroll
  for (int r = 0; r < 8; ++r) {
    int m = r + lh * 8;
    yp[(size_t)m * DIM + lm]      = c0[r];
    yp[(size_t)m * DIM + 16 + lm] = c1[r];
  }
}

// tiny final projection 32 -> 2 (memory bound; VALU is fine)
__global__ void gemm_small2(const float* __restrict__ X, const float* __restrict__ W,
                            float* __restrict__ Y, int n) {
  int i = blockIdx.x * blockDim.x + threadIdx.x;
  if (i >= n) return;
  const float* x = X + (size_t)i * DIM;
  float a0 = 0.f, a1 = 0.f;
#pragma unroll
  for (int k = 0; k < DIM; ++k) {
    float xv = x[k];
    a0 = fmaf(xv, W[k * 2 + 0], a0);
    a1 = fmaf(xv, W[k * 2 + 1], a1);
  }
  Y[(size_t)i * 2 + 0] = a0;
  Y[(size_t)i * 2 + 1] = a1;
}

// ---------------- sparse: GCN message scatter ----------------

// Y[dst*C + c] += H[src*C + c] * dinv[src]*dinv[dst]   (thread per edge-column)
template <int C>
__global__ void scatter_edges(float* __restrict__ Y, const float* __restrict__ H,
                              const int* __restrict__ src, const int* __restrict__ dst,
                              const float* __restrict__ dinv, int E_) {
  long long t = (long long)blockIdx.x * blockDim.x + threadIdx.x;
  if (t >= (long long)E_ * C) return;
  int e = (int)(t / C);
  int c = (int)(t % C);
  int s = src[e];
  int d = dst[e];
  float w = dinv[s] * dinv[d];
  atomicAdd(Y + (size_t)d * C + c, H[(size_t)s * C + c] * w);
}

// Y[i*C+c] += H[i*C+c]*dinv[i]^2 + bias[c]; optional ReLU
template <int C, bool RELU>
__global__ void selfloop_bias(float* __restrict__ Y, const float* __restrict__ H,
                              const float* __restrict__ dinv, const float* __restrict__ bias,
                              int n) {
  long long t = (long long)blockIdx.x * blockDim.x + threadIdx.x;
  if (t >= (long long)n * C) return;
  int i = (int)(t / C);
  int c = (int)(t % C);
  float di = dinv[i];
  float v = Y[(size_t)i * C + c] + H[(size_t)i * C + c] * di * di + bias[c];
  if (RELU) v = fmaxf(v, 0.0f);
  Y[(size_t)i * C + c] = v;
}

// T[sidx[e]*32+c] += F[gidx[e]*32+c]  (scatter_mean numerator)
__global__ void scatter_feat32(float* __restrict__ T, const float* __restrict__ F,
                               const int* __restrict__ gidx, const int* __restrict__ sidx,
                               int E_) {
  long long t = (long long)blockIdx.x * blockDim.x + threadIdx.x;
  if (t >= (long long)E_ * DIM) return;
  int e = (int)(t / DIM);
  int c = (int)(t % DIM);
  atomicAdd(T + (size_t)sidx[e] * DIM + c, F[(size_t)gidx[e] * DIM + c]);
}

// B[i*32+c] += T[i*32+c] / max(cnt[i], 1)
__global__ void add_mean32(float* __restrict__ B, const float* __restrict__ T,
                           const float* __restrict__ cnt, int n) {
  long long t = (long long)blockIdx.x * blockDim.x + threadIdx.x;
  if (t >= (long long)n * DIM) return;
  int i = (int)(t / DIM);
  int c = (int)(t % DIM);
  B[(size_t)i * DIM + c] += T[(size_t)i * DIM + c] / fmaxf(cnt[i], 1.0f);
}

// ---------------- host ----------------

static inline int cdiv_ll(long long a, int b) { return (int)((a + b - 1) / b); }

extern "C" void kernel_launch(void* const* d_in, const int* in_sizes, int n_in,
                              void* d_out, int out_size, void* d_ws, size_t ws_size,
                              hipStream_t stream) {
  // setup_inputs() dict order:
  // 0:x0 1:ei0 2:x1 3:ei1 4:x2 5:ei2 6:le1 7:le2
  // 8:Wl0 9:bl0 10:Wo0 11:bo0 12:Wl1 13:bl1 14:Wo1 15:bo1 16:Wl2 17:bl2 18:Wo2 19:bo2
  // 20:Wp1 21:bp1 22:Wp2 23:bp2
  const float* x1  = (const float*)d_in[2];
  const int*   ei1 = (const int*)d_in[3];
  const float* x2  = (const float*)d_in[4];
  const int*   ei2 = (const int*)d_in[5];
  const int*   le2 = (const int*)d_in[7];
  const float* Wl1 = (const float*)d_in[12]; const float* bl1 = (const float*)d_in[13];
  const float* Wl2 = (const float*)d_in[16]; const float* bl2 = (const float*)d_in[17];
  const float* Wo2 = (const float*)d_in[18]; const float* bo2 = (const float*)d_in[19];
  const float* Wp1 = (const float*)d_in[20]; const float* bp1 = (const float*)d_in[21];
  const float* Wp2 = (const float*)d_in[22]; const float* bp2 = (const float*)d_in[23];
  float* out = (float*)d_out;

  const int Nn = in_sizes[2] / FIN;  // 200000
  const int Eg = in_sizes[5] / 2;    // 2000000 intra-layer edges
  const int El = in_sizes[7] / 2;    // 500000 cross-layer edges

  // workspace carve-up (floats): T,A,B,C (N*32 each) + dinv1,dinv2,cnt (N) + T2 (N*2)
  float* ws    = (float*)d_ws;
  float* T     = ws;
  float* A     = T  + (size_t)Nn * DIM;
  float* B     = A  + (size_t)Nn * DIM;
  float* Cb    = B  + (size_t)Nn * DIM;
  float* dinv1 = Cb + (size_t)Nn * DIM;
  float* dinv2 = dinv1 + Nn;
  float* cnt   = dinv2 + Nn;
  float* T2    = cnt + Nn;

  const int TB = 256;
  const long long NC  = (long long)Nn * DIM;
  const long long EC  = (long long)Eg * DIM;
  const long long LC  = (long long)El * DIM;
  const int waveTiles = (Nn + 15) / 16;           // 16 rows per wave
  const int gemmGrid  = (waveTiles + 7) / 8;      // 8 waves per 256-thread block

  const int* e1s = ei1;       const int* e1d = ei1 + Eg;
  const int* e2s = ei2;       const int* e2d = ei2 + Eg;
  const int* l2i = le2;       const int* l2o = le2 + El;   // le2[0], le2[1]

  // degrees -> dinv for both intra-layer edge lists
  zero_f32<<<cdiv_ll(Nn, TB), TB, 0, stream>>>(dinv1, Nn);
  zero_f32<<<cdiv_ll(Nn, TB), TB, 0, stream>>>(dinv2, Nn);
  count_idx<<<cdiv_ll(Eg, TB), TB, 0, stream>>>(dinv1, e1d, Eg);
  count_idx<<<cdiv_ll(Eg, TB), TB, 0, stream>>>(dinv2, e2d, Eg);
  finalize_dinv<<<cdiv_ll(Nn, TB), TB, 0, stream>>>(dinv1, Nn);
  finalize_dinv<<<cdiv_ll(Nn, TB), TB, 0, stream>>>(dinv2, Nn);

  // A = A2 = relu(gcn(x2, Wl2, bl2, ei2))
  gemm_wmma_k<FIN><<<gemmGrid, TB, 0, stream>>>(x2, Wl2, T, Nn);
  zero_f32<<<cdiv_ll(NC, TB), TB, 0, stream>>>(A, NC);
  scatter_edges<DIM><<<cdiv_ll(EC, TB), TB, 0, stream>>>(A, T, e2s, e2d, dinv2, Eg);
  selfloop_bias<DIM, true><<<cdiv_ll(NC, TB), TB, 0, stream>>>(A, T, dinv2, bl2, Nn);

  // B = A1 = relu(gcn(x1, Wl1, bl1, ei1))
  gemm_wmma_k<FIN><<<gemmGrid, TB, 0, stream>>>(x1, Wl1, T, Nn);
  zero_f32<<<cdiv_ll(NC, TB), TB, 0, stream>>>(B, NC);
  scatter_edges<DIM><<<cdiv_ll(EC, TB), TB, 0, stream>>>(B, T, e1s, e1d, dinv1, Eg);
  selfloop_bias<DIM, true><<<cdiv_ll(NC, TB), TB, 0, stream>>>(B, T, dinv1, bl1, Nn);

  // B = L1 = B + scatter_mean(A[le2[1]] -> le2[0])
  zero_f32<<<cdiv_ll(NC, TB), TB, 0, stream>>>(T, NC);
  zero_f32<<<cdiv_ll(Nn, TB), TB, 0, stream>>>(cnt, Nn);
  scatter_feat32<<<cdiv_ll(LC, TB), TB, 0, stream>>>(T, A, l2o, l2i, El);
  count_idx<<<cdiv_ll(El, TB), TB, 0, stream>>>(cnt, l2i, El);
  add_mean32<<<cdiv_ll(NC, TB), TB, 0, stream>>>(B, T, cnt, Nn);

  // C = LF2 = relu(gcn(A, Wo2, bo2, ei2))
  gemm_wmma_k<DIM><<<gemmGrid, TB, 0, stream>>>(A, Wo2, T, Nn);
  zero_f32<<<cdiv_ll(NC, TB), TB, 0, stream>>>(Cb, NC);
  scatter_edges<DIM><<<cdiv_ll(EC, TB), TB, 0, stream>>>(Cb, T, e2s, e2d, dinv2, Eg);
  selfloop_bias<DIM, true><<<cdiv_ll(NC, TB), TB, 0, stream>>>(Cb, T, dinv2, bo2, Nn);

  // C = F = C + scatter_mean(B[le2[0]] -> le2[1])
  zero_f32<<<cdiv_ll(NC, TB), TB, 0, stream>>>(T, NC);
  zero_f32<<<cdiv_ll(Nn, TB), TB, 0, stream>>>(cnt, Nn);
  scatter_feat32<<<cdiv_ll(LC, TB), TB, 0, stream>>>(T, B, l2i, l2o, El);
  count_idx<<<cdiv_ll(El, TB), TB, 0, stream>>>(cnt, l2o, El);
  add_mean32<<<cdiv_ll(NC, TB), TB, 0, stream>>>(Cb, T, cnt, Nn);

  // A (reuse) = H = relu(gcn(C, Wp1, bp1, ei2))
  gemm_wmma_k<DIM><<<gemmGrid, TB, 0, stream>>>(Cb, Wp1, T, Nn);
  zero_f32<<<cdiv_ll(NC, TB), TB, 0, stream>>>(A, NC);
  scatter_edges<DIM><<<cdiv_ll(EC, TB), TB, 0, stream>>>(A, T, e2s, e2d, dinv2, Eg);
  selfloop_bias<DIM, true><<<cdiv_ll(NC, TB), TB, 0, stream>>>(A, T, dinv2, bp1, Nn);

  // out = gcn(A, Wp2, bp2, ei2)  [N x 2], no relu
  gemm_small2<<<cdiv_ll(Nn, TB), TB, 0, stream>>>(A, Wp2, T2, Nn);
  zero_f32<<<cdiv_ll((long long)Nn * 2, TB), TB, 0, stream>>>(out, (long long)Nn * 2);
  scatter_edges<2><<<cdiv_ll((long long)Eg * 2, TB), TB, 0, stream>>>(out, T2, e2s, e2d, dinv2, Eg);
  selfloop_bias<2, false><<<cdiv_ll((long long)Nn * 2, TB), TB, 0, stream>>>(out, T2, dinv2, bp2, Nn);
}